// GAT_83906481094720
// MI455X (gfx1250) — compile-verified
//
#include <hip/hip_runtime.h>
#include <math.h>

// ---------- problem constants ----------
#define BB 64
#define NO 4000
#define NM 128
#define DD 64
#define NCHUNK 4
#define OCH (NO / NCHUNK)   // 1000 opes per chunk
#define SS 40               // opes per LDS super-step
#define NSS (OCH / SS)      // 25 super-steps

typedef __attribute__((ext_vector_type(2))) float v2f;
typedef __attribute__((ext_vector_type(8))) float v8f;
typedef unsigned int uint;
typedef __attribute__((ext_vector_type(4))) uint v4u;
typedef __attribute__((ext_vector_type(8))) uint v8u;

__device__ __forceinline__ v8f wmma4(v2f a, v2f b, v8f c) {
  // D(16x16,f32) = A(16x4,f32) x B(4x16,f32) + C
  return __builtin_amdgcn_wmma_f32_16x16x4_f32(false, a, false, b, (short)0, c,
                                               false, false);
}

// ---------- TDM: async 2D tile DMA global -> LDS (Tensor Data Mover) ----------
// Builds D# group0 (4 SGPRs) + group1 (8 SGPRs) for a 2D tile of 4-byte
// elements: tile_x columns x tile_y rows, row stride stride_x elements.
// 2D tensors need only groups 0/1 (VADDR2/3 = NULL form).
__device__ __forceinline__ void tdm_load_2d(void* lds_dst, const void* gsrc,
                                            uint tile_x, uint tile_y,
                                            uint stride_x) {
  unsigned long long ga = (unsigned long long)(uintptr_t)gsrc;
  uint lds = (uint)(uintptr_t)lds_dst;  // generic LDS addr: low 32 bits = LDS offset
  v4u g0;
  g0.x = 1u;                                   // count=1 valid descriptor
  g0.y = lds;                                  // lds_addr [63:32]
  g0.z = (uint)ga;                             // global_addr [95:64]
  g0.w = ((uint)(ga >> 32) & 0x01FFFFFFu) | (2u << 30);  // ga[56:32] | type=2
  v8u g1;
  g1[0] = 2u << 16;                            // wg_mask=0, data_size=2 (4B)
  g1[1] = (tile_x & 0xFFFFu) << 16;            // tensor_dim0[15:0]
  g1[2] = (tile_x >> 16) | ((tile_y & 0xFFFFu) << 16);  // td0 hi | td1 lo
  g1[3] = (tile_y >> 16) | ((tile_x & 0xFFFFu) << 16);  // td1 hi | tile_dim0
  g1[4] = tile_y & 0xFFFFu;                    // tile_dim1, tile_dim2=0
  g1[5] = stride_x;                            // tensor_dim0_stride [31:0]
  g1[6] = 0u;
  g1[7] = 0u;
  asm volatile("tensor_load_to_lds %0, %1" : : "s"(g0), "s"(g1) : "memory");
}

// ---------- kernel 1: fold alpha vectors into W: weff[d] = sum_k alpha[k]*W[k,d] ----------
__global__ __launch_bounds__(64) void prep_kernel(
    const float* __restrict__ W_ope, const float* __restrict__ W_mac,
    const float* __restrict__ alpha_ope, const float* __restrict__ alpha_mac,
    float* __restrict__ weff_ope, float* __restrict__ weff_mac) {
  int d = threadIdx.x;
  float s1 = 0.f, s2 = 0.f;
#pragma unroll
  for (int k = 0; k < DD; ++k) {
    s1 += alpha_ope[k] * W_ope[k * DD + d];
    s2 += alpha_mac[k] * W_mac[k * DD + d];
  }
  weff_ope[d] = s1;
  weff_mac[d] = s2;
}

// ---------- kernel 2: out[row] = feat[row,:] . w  (wave32 per row) ----------
__global__ __launch_bounds__(256) void rowdot_kernel(
    const float* __restrict__ feat, const float* __restrict__ w,
    float* __restrict__ out, int nrows) {
  int gid = blockIdx.x * blockDim.x + threadIdx.x;
  int row = gid >> 5;
  int lane = threadIdx.x & 31;
  if (row >= nrows) return;
  const float2 f = ((const float2*)(feat + (size_t)row * DD))[lane];
  const float2 wv = ((const float2*)w)[lane];
  float v = f.x * wv.x + f.y * wv.y;
#pragma unroll
  for (int off = 16; off > 0; off >>= 1) v += __shfl_xor(v, off, 32);
  if (lane == 0) out[row] = v;
}

// ---------- kernel 3: streaming masked-softmax attention contraction ----------
// grid = BB * NCHUNK blocks, 256 threads (8 waves).
// wave w owns output rows m in [16w,16w+16), columns d in [0,64).
// Rp[m,d] = sum_o alpha_unnorm[o,m]*feat_ope[o,d], sump[m] = sum_o alpha_unnorm,
// alpha_unnorm = proc*exp(leaky(a_ope[o]+a_mac[m])) (proc in {0,1} folds the mask).
// proc tiles stream through LDS via async TDM DMA; feat tiles are staged
// pair-interleaved so every WMMA B operand is a single aligned ds_load_b64.
__global__ __launch_bounds__(256) void attn_kernel(
    const float* __restrict__ proc, const float* __restrict__ feat_ope,
    const float* __restrict__ a_ope, const float* __restrict__ a_mac,
    float* __restrict__ Rp, float* __restrict__ sump) {
  __shared__ float procT[2][SS * NM];          // [o][m] tile (TDM dest), 2x20KB
  __shared__ v2f featP[2][(SS / 2) * DD];      // pair-interleaved feat tile
  __shared__ float aopeT[2][SS];

  const int bc = blockIdx.x;
  const int b = bc / NCHUNK;
  const int c = bc % NCHUNK;
  const int tid = threadIdx.x;
  const int wave = tid >> 5;
  const int lane = tid & 31;
  const int ml = lane & 15;       // N column within 16-wide tile / A row
  const int hi = lane >> 4;       // 0: K={0,1}, 1: K={2,3} (ISA A/B layout)
  const int mloc = (wave << 4) | ml;

  const float amac = a_mac[b * NM + mloc];
  const float* procB = proc + ((size_t)b * NO + (size_t)c * OCH) * NM;
  const float* featB = feat_ope + ((size_t)b * NO + (size_t)c * OCH) * DD;
  const float* aopeB = a_ope + (size_t)b * NO + (size_t)c * OCH;

  v8f acc0 = {}, acc1 = {}, acc2 = {}, acc3 = {};
  float ssum = 0.f;

  // cooperative feat tile load: featP[op][d] = {feat[2op][d], feat[2op+1][d]}
  auto load_feat = [&](int buf, int s) {
    const float* src = featB + (size_t)s * SS * DD;
#pragma unroll
    for (int i = 0; i < (SS / 2) * DD / 256; ++i) {  // 5 elems per thread
      int e = tid + i * 256;
      int op = e >> 6;          // pair index
      int d = e & 63;
      const float* g = src + (size_t)(2 * op) * DD + d;
      v2f v;
      v.x = g[0];
      v.y = g[DD];
      featP[buf][op * DD + d] = v;  // ds_store_b64, conflict-free
    }
    if (tid < SS) aopeT[buf][tid] = aopeB[s * SS + tid];
    // prefetch next feat tile region into cache hierarchy
    const char* pn = (const char*)(src + SS * DD);
    if (s + 1 < NSS && tid < (SS * DD * 4) / 128)
      __builtin_prefetch(pn + tid * 128, 0, 1);
  };

  // prologue: async-DMA proc tile 0, stage feat tile 0
  if (wave == 0)
    tdm_load_2d(&procT[0][0], procB, NM, SS, NM);
  load_feat(0, 0);
  if (wave == 0) __builtin_amdgcn_s_wait_tensorcnt(0);
  __syncthreads();

  for (int s = 0; s < NSS; ++s) {
    const int buf = s & 1;
    if (s + 1 < NSS) {
      if (wave == 0)
        tdm_load_2d(&procT[buf ^ 1][0], procB + (size_t)(s + 1) * SS * NM, NM,
                    SS, NM);
      load_feat(buf ^ 1, s + 1);
    }
#pragma unroll
    for (int k = 0; k < SS / 4; ++k) {
      const int kk = 4 * k + (hi << 1);     // this lane's first K (o) index
      const v2f a01 = *(const v2f*)&aopeT[buf][kk];          // ds_load_b64
      const float p0 = procT[buf][kk * NM + mloc];
      const float p1 = procT[buf][kk * NM + NM + mloc];
      float t0 = a01.x + amac, t1 = a01.y + amac;
      t0 = fmaxf(t0, 0.2f * t0);            // leaky_relu, slope 0.2
      t1 = fmaxf(t1, 0.2f * t1);
      v2f A;
      A.x = p0 * __expf(t0);                // proc in {0,1}: mask folds in
      A.y = p1 * __expf(t1);
      ssum += A.x + A.y;
      const v2f* fp = &featP[buf][(kk >> 1) * DD + ml];
      v2f B0 = fp[0];                       // aligned ds_load_b64 each
      v2f B1 = fp[16];
      v2f B2 = fp[32];
      v2f B3 = fp[48];
      acc0 = wmma4(A, B0, acc0);
      acc1 = wmma4(A, B1, acc1);
      acc2 = wmma4(A, B2, acc2);
      acc3 = wmma4(A, B3, acc3);
    }
    if (wave == 0) __builtin_amdgcn_s_wait_tensorcnt(0);
    __syncthreads();
  }

  // per-m sum: lanes l and l+16 hold halves for the same m
  float ssum2 = ssum + __shfl_xor(ssum, 16, 32);
  if (lane < 16) sump[(size_t)(b * NCHUNK + c) * NM + mloc] = ssum2;

  // D layout: VGPR j holds M=j (lanes 0-15) / M=j+8 (lanes 16-31), N = ml
  float* rp = Rp + (size_t)(b * NCHUNK + c) * NM * DD;
#pragma unroll
  for (int j = 0; j < 8; ++j) {
    const int m = (wave << 4) + j + 8 * hi;
    rp[m * DD + 0  + ml] = acc0[j];
    rp[m * DD + 16 + ml] = acc1[j];
    rp[m * DD + 32 + ml] = acc2[j];
    rp[m * DD + 48 + ml] = acc3[j];
  }
}

// ---------- kernel 4: combine partials, normalize, out = Rn@Wo^T + feat_mac@Wm^T ----------
__global__ __launch_bounds__(256) void combine_kernel(
    const float* __restrict__ Rp, const float* __restrict__ sump,
    const float* __restrict__ feat_mac, const float* __restrict__ W_ope,
    const float* __restrict__ W_mac, float* __restrict__ out) {
  __shared__ float Rn[NM * DD];
  __shared__ float FM[NM * DD];
  __shared__ float WO[DD * DD];
  __shared__ float WM[DD * DD];
  __shared__ float stot[NM];

  const int b = blockIdx.x;
  const int tid = threadIdx.x;

  if (tid < NM) {
    float s = 0.f;
#pragma unroll
    for (int c = 0; c < NCHUNK; ++c)
      s += sump[(size_t)(b * NCHUNK + c) * NM + tid];
    stot[tid] = (s > 0.f) ? (1.f / s) : 0.f;
  }
  __syncthreads();

  for (int i = tid; i < NM * DD; i += 256) {
    float v = 0.f;
#pragma unroll
    for (int c = 0; c < NCHUNK; ++c)
      v += Rp[(size_t)(b * NCHUNK + c) * NM * DD + i];
    Rn[i] = v * stot[i >> 6];
    FM[i] = feat_mac[(size_t)b * NM * DD + i];
  }
  for (int i = tid; i < DD * DD; i += 256) {
    WO[i] = W_ope[i];
    WM[i] = W_mac[i];
  }
  __syncthreads();

  const int wave = tid >> 5;
  const int lane = tid & 31;
  const int ml = lane & 15;
  const int hi = lane >> 4;
  const int arow = (wave << 4) + ml;

  v8f acc0 = {}, acc1 = {}, acc2 = {}, acc3 = {};

  // out[m,n] = sum_K Rn[m,K]*Wo[n,K] + sum_K FM[m,K]*Wm[n,K]
#pragma unroll
  for (int k4 = 0; k4 < DD / 4; ++k4) {
    const int K = 4 * k4 + 2 * hi;
    v2f A = *(const v2f*)&Rn[arow * DD + K];
    v2f B0 = *(const v2f*)&WO[(0 * 16 + ml) * DD + K];
    v2f B1 = *(const v2f*)&WO[(1 * 16 + ml) * DD + K];
    v2f B2 = *(const v2f*)&WO[(2 * 16 + ml) * DD + K];
    v2f B3 = *(const v2f*)&WO[(3 * 16 + ml) * DD + K];
    acc0 = wmma4(A, B0, acc0);
    acc1 = wmma4(A, B1, acc1);
    acc2 = wmma4(A, B2, acc2);
    acc3 = wmma4(A, B3, acc3);
  }
#pragma unroll
  for (int k4 = 0; k4 < DD / 4; ++k4) {
    const int K = 4 * k4 + 2 * hi;
    v2f A = *(const v2f*)&FM[arow * DD + K];
    v2f B0 = *(const v2f*)&WM[(0 * 16 + ml) * DD + K];
    v2f B1 = *(const v2f*)&WM[(1 * 16 + ml) * DD + K];
    v2f B2 = *(const v2f*)&WM[(2 * 16 + ml) * DD + K];
    v2f B3 = *(const v2f*)&WM[(3 * 16 + ml) * DD + K];
    acc0 = wmma4(A, B0, acc0);
    acc1 = wmma4(A, B1, acc1);
    acc2 = wmma4(A, B2, acc2);
    acc3 = wmma4(A, B3, acc3);
  }

  float* outB = out + (size_t)b * NM * DD;
#pragma unroll
  for (int j = 0; j < 8; ++j) {
    const int m = (wave << 4) + j + 8 * hi;
    outB[m * DD + 0  + ml] = acc0[j];
    outB[m * DD + 16 + ml] = acc1[j];
    outB[m * DD + 32 + ml] = acc2[j];
    outB[m * DD + 48 + ml] = acc3[j];
  }
}

// ---------- host launcher ----------
extern "C" void kernel_launch(void* const* d_in, const int* in_sizes, int n_in,
                              void* d_out, int out_size, void* d_ws,
                              size_t ws_size, hipStream_t stream) {
  const float* proc = (const float*)d_in[0];      // [B,O,M]
  // d_in[1] = batch_idxes (arange, identity) -- unused
  const float* feat_ope = (const float*)d_in[2];  // [B,O,D]
  const float* feat_mac = (const float*)d_in[3];  // [B,M,D]
  const float* W_ope = (const float*)d_in[4];     // [D,D]
  const float* W_mac = (const float*)d_in[5];     // [D,D]
  const float* alpha_ope = (const float*)d_in[6]; // [D]
  const float* alpha_mac = (const float*)d_in[7]; // [D]
  float* out = (float*)d_out;                     // [B,M,D]

  float* ws = (float*)d_ws;
  float* weff_ope = ws;                               // 64
  float* weff_mac = weff_ope + DD;                    // 64
  float* a_ope = weff_mac + DD;                       // B*O
  float* a_mac = a_ope + (size_t)BB * NO;             // B*M
  float* sump = a_mac + (size_t)BB * NM;              // B*NCHUNK*M
  float* Rp = sump + (size_t)BB * NCHUNK * NM;        // B*NCHUNK*M*D

  prep_kernel<<<1, 64, 0, stream>>>(W_ope, W_mac, alpha_ope, alpha_mac,
                                    weff_ope, weff_mac);

  {
    int rows = BB * NO;                 // 256000 rows, 8 waves per block
    rowdot_kernel<<<rows / 8, 256, 0, stream>>>(feat_ope, weff_ope, a_ope, rows);
  }
  {
    int rows = BB * NM;                 // 8192 rows
    rowdot_kernel<<<rows / 8, 256, 0, stream>>>(feat_mac, weff_mac, a_mac, rows);
  }

  attn_kernel<<<BB * NCHUNK, 256, 0, stream>>>(proc, feat_ope, a_ope, a_mac,
                                               Rp, sump);

  combine_kernel<<<BB, 256, 0, stream>>>(Rp, sump, feat_mac, W_ope, W_mac, out);
}